// SelectiveScan_137438954197
// MI455X (gfx1250) — compile-verified
//
#include <hip/hip_runtime.h>
#include <hip/hip_bf16.h>

// Shapes from the reference: B=4, K=2, D=192, L=4096, N=16, R=12, C=R+2N=44
constexpr int NB  = 4;
constexpr int NK  = 2;
constexpr int ND  = 192;
constexpr int NL  = 4096;
constexpr int NN  = 16;
constexpr int NR  = 12;
constexpr int NC  = 44;   // R + 2N rows of x_dbl
constexpr int NCP = 48;   // padded to multiple of 16 for WMMA tiling

// Set to 0 if the async-store inline asm fails to assemble.
#ifndef USE_ASYNC_STORE
#define USE_ASYNC_STORE 1
#endif

typedef __attribute__((ext_vector_type(2))) float v2f;
typedef __attribute__((ext_vector_type(8))) float v8f;

// ---------------------------------------------------------------------------
// Kernel 1: x_dbl[bk, c, l] = sum_d W[k, c, d] * x[bk, d, l]
// fp32 WMMA (V_WMMA_F32_16X16X4_F32), one wave32 per 16x16 (c x l) tile.
// (unchanged from round 1 — confirmed v_wmma lowering)
// ---------------------------------------------------------------------------
__global__ __launch_bounds__(32)
void xdbl_wmma_kernel(const float* __restrict__ x,
                      const float* __restrict__ W,
                      float* __restrict__ xdbl) {
  const int l0   = blockIdx.x * 16;
  const int c0   = blockIdx.y * 16;
  const int bk   = blockIdx.z;
  const int k    = bk & 1;
  const int lane = threadIdx.x;
  const int half = lane >> 4;
  const int m    = lane & 15;
  const int kb   = half * 2;

  const float* xrow = x + (size_t)bk * ND * NL;
  const float* Wk   = W + (size_t)k * NC * ND;

  const int  c     = c0 + m;
  const bool cok   = (c < NC);
  const int  csafe = cok ? c : 0;
  const float ascale = cok ? 1.0f : 0.0f;

  v8f acc = {0.f, 0.f, 0.f, 0.f, 0.f, 0.f, 0.f, 0.f};

  for (int d0 = 0; d0 < ND; d0 += 4) {
    v2f a = *(const v2f*)(Wk + (size_t)csafe * ND + d0 + kb);
    a.x *= ascale;
    a.y *= ascale;
    v2f bm;
    bm.x = xrow[(size_t)(d0 + kb)     * NL + l0 + m];
    bm.y = xrow[(size_t)(d0 + kb + 1) * NL + l0 + m];
    acc = __builtin_amdgcn_wmma_f32_16x16x4_f32(
        false, a, false, bm, (short)0, acc, false, false);
  }

  float* outp = xdbl + (size_t)bk * NCP * NL;
#pragma unroll
  for (int i = 0; i < 8; ++i) {
    const int row = c0 + i + half * 8;
    outp[(size_t)row * NL + l0 + m] = acc[i];
  }
}

// ---------------------------------------------------------------------------
// Kernel 2: delta[bk, d, l] = softplus( sum_r dtw[k,d,r]*dts[bk,r,l] + bias )
// (unchanged from round 1)
// ---------------------------------------------------------------------------
__global__ __launch_bounds__(32)
void delta_wmma_kernel(const float* __restrict__ xdbl,
                       const float* __restrict__ dtw,
                       const float* __restrict__ bias,
                       float* __restrict__ delta) {
  const int l0   = blockIdx.x * 16;
  const int d0   = blockIdx.y * 16;
  const int bk   = blockIdx.z;
  const int k    = bk & 1;
  const int lane = threadIdx.x;
  const int half = lane >> 4;
  const int m    = lane & 15;
  const int kb   = half * 2;

  const float* dts = xdbl + (size_t)bk * NCP * NL;
  const float* Wk  = dtw + (size_t)k * ND * NR;

  v8f acc = {0.f, 0.f, 0.f, 0.f, 0.f, 0.f, 0.f, 0.f};

#pragma unroll
  for (int r0 = 0; r0 < NR; r0 += 4) {
    v2f a = *(const v2f*)(Wk + (size_t)(d0 + m) * NR + r0 + kb);
    v2f bm;
    bm.x = dts[(size_t)(r0 + kb)     * NL + l0 + m];
    bm.y = dts[(size_t)(r0 + kb + 1) * NL + l0 + m];
    acc = __builtin_amdgcn_wmma_f32_16x16x4_f32(
        false, a, false, bm, (short)0, acc, false, false);
  }

#pragma unroll
  for (int i = 0; i < 8; ++i) {
    const int row = d0 + i + half * 8;
    const float z = acc[i] + bias[k * ND + row];
    const float sp = (z > 20.0f) ? z : __logf(1.0f + __expf(z));
    delta[((size_t)bk * ND + row) * NL + l0 + m] = sp;
  }
}

// ---------------------------------------------------------------------------
// Kernel 2b: transpose B/C rows of x_dbl into BC_t[bk][l][32]
// (n-contiguous: j=0..15 -> B[n], j=16..31 -> C[n]) so the scan can fetch all
// 32 coefficients for one l as a single 128B line (float4 x8).
// Classic 32x32 LDS tile transpose, padded stride 33 (conflict-free).
// ---------------------------------------------------------------------------
__global__ __launch_bounds__(256)
void bct_transpose_kernel(const float* __restrict__ xdbl,
                          float* __restrict__ bct) {
  __shared__ float tile[32][33];
  const int l0 = blockIdx.x * 32;
  const int bk = blockIdx.y;
  const int tx = threadIdx.x & 31;
  const int ty = threadIdx.x >> 5;   // 0..7

  const float* src = xdbl + ((size_t)bk * NCP + NR) * NL;  // rows 12..43 = B,C
#pragma unroll
  for (int r = 0; r < 4; ++r) {
    const int j = ty + 8 * r;                   // 0..31
    tile[j][tx] = src[(size_t)j * NL + l0 + tx];  // coalesced read
  }
  __syncthreads();
  float* dst = bct + (size_t)bk * NL * 32;
#pragma unroll
  for (int r = 0; r < 4; ++r) {
    const int lr = ty + 8 * r;
    dst[(size_t)(l0 + lr) * 32 + tx] = tile[tx][lr];  // coalesced write
  }
}

// ---------------------------------------------------------------------------
// Kernel 3: chunked two-pass selective scan.
// Block = 8 waves = 8 d-rows of one (b,k) (shared BC_t lines stay hot in WGP$).
// Each wave owns one row; each lane owns a contiguous 128-element L-chunk with
// all 16 states in registers.
//   Pass 1:  per-chunk (P_n = prod dA, H_n = local scan end state from h=0)
//   Prefix:  5-step __shfl_up Hillis-Steele scan with (a1*a2, a2*b1+b2),
//            shifted to exclusive -> state entering each chunk
//   Pass 2:  replay chunk from correct state, y_l = sum_n h_n*C_n + D*x_l
// y goes through a padded LDS tile and is flushed with coalesced
// global_store_async_from_lds_b32 (ASYNCcnt) when USE_ASYNC_STORE=1.
// ---------------------------------------------------------------------------
__global__ __launch_bounds__(256)
void scan_chunked_kernel(const float* __restrict__ x,
                         const float* __restrict__ delta,
                         const float* __restrict__ bct,
                         const float* __restrict__ A_logs,
                         const float* __restrict__ Ds,
                         float* __restrict__ y) {
  constexpr int CH = NL / 32;                 // 128 l per lane
  __shared__ float ytile[8][32 * 33];         // 33.8 KB, stride-33 pad

  const int tid = threadIdx.x;
  const int la  = tid & 31;
  const int w   = tid >> 5;
  const int bk  = blockIdx.x / (ND / 8);
  const int d   = (blockIdx.x % (ND / 8)) * 8 + w;
  const int k   = bk & 1;

  const float* xrow = x     + ((size_t)bk * ND + d) * NL;
  const float* drow = delta + ((size_t)bk * ND + d) * NL;
  const float* bcb  = bct   + (size_t)bk * NL * 32;
  float*       yrow = y     + ((size_t)bk * ND + d) * NL;

  float A[16];
#pragma unroll
  for (int n = 0; n < 16; ++n)
    A[n] = -__expf(A_logs[((size_t)k * ND + d) * NN + n]);
  const float Dval = Ds[k * ND + d];

  const int l0 = la * CH;

  // ---------------- Pass 1 ----------------
  float P[16], H[16];
#pragma unroll
  for (int n = 0; n < 16; ++n) { P[n] = 1.0f; H[n] = 0.0f; }

  for (int i = 0; i < CH; ++i) {
    const int l = l0 + i;
    const float dl = drow[l];
    const float xv = xrow[l];
    const float du = dl * xv;
    const float4* bc = (const float4*)(bcb + (size_t)l * 32);
    __builtin_prefetch(bcb + (size_t)((l + 8 < NL) ? (l + 8) : l) * 32, 0, 0);
    float Bv[16];
#pragma unroll
    for (int j = 0; j < 4; ++j) {
      const float4 t = bc[j];
      Bv[4*j+0] = t.x; Bv[4*j+1] = t.y; Bv[4*j+2] = t.z; Bv[4*j+3] = t.w;
    }
#pragma unroll
    for (int n = 0; n < 16; ++n) {
      const float dA = __expf(dl * A[n]);     // independent -> pipelined
      H[n] = fmaf(dA, H[n], du * Bv[n]);
      P[n] *= dA;
    }
  }

  // ---------------- wave-wide prefix ----------------
#pragma unroll
  for (int off = 1; off < 32; off <<= 1) {
#pragma unroll
    for (int n = 0; n < 16; ++n) {
      const float Pp = __shfl_up(P[n], off);
      const float Hp = __shfl_up(H[n], off);
      const float Hn = fmaf(P[n], Hp, H[n]);
      const float Pn = P[n] * Pp;
      H[n] = (la >= off) ? Hn : H[n];
      P[n] = (la >= off) ? Pn : P[n];
    }
  }
  float h[16];
#pragma unroll
  for (int n = 0; n < 16; ++n) {
    const float He = __shfl_up(H[n], 1);
    h[n] = (la == 0) ? 0.0f : He;             // state entering this chunk
  }

  // ---------------- Pass 2 ----------------
  float* myt = &ytile[w][0];
  for (int ib = 0; ib < CH / 32; ++ib) {
    for (int ii = 0; ii < 32; ++ii) {
      const int i = ib * 32 + ii;
      const int l = l0 + i;
      const float dl = drow[l];
      const float xv = xrow[l];
      const float du = dl * xv;
      const float4* bc = (const float4*)(bcb + (size_t)l * 32);
      __builtin_prefetch(bcb + (size_t)((l + 8 < NL) ? (l + 8) : l) * 32, 0, 0);
      float Bv[16], Cv[16];
#pragma unroll
      for (int j = 0; j < 4; ++j) {
        const float4 tb = bc[j];
        const float4 tc = bc[j + 4];
        Bv[4*j+0] = tb.x; Bv[4*j+1] = tb.y; Bv[4*j+2] = tb.z; Bv[4*j+3] = tb.w;
        Cv[4*j+0] = tc.x; Cv[4*j+1] = tc.y; Cv[4*j+2] = tc.z; Cv[4*j+3] = tc.w;
      }
      float acc = 0.0f;
#pragma unroll
      for (int n = 0; n < 16; ++n) {
        const float dA = __expf(dl * A[n]);
        h[n] = fmaf(dA, h[n], du * Bv[n]);
        acc  = fmaf(h[n], Cv[n], acc);
      }
      myt[la * 33 + ii] = fmaf(Dval, xv, acc);  // stride-33: conflict-free
    }
    __syncthreads();
#if USE_ASYNC_STORE
    {
      asm volatile("s_wait_dscnt 0x0" ::: "memory");
#pragma unroll 1
      for (int jj = 0; jj < 32; ++jj) {
        const unsigned lds_off = (unsigned)(uintptr_t)(&ytile[w][jj * 33 + la]);
        float* gp = yrow + (size_t)jj * CH + ib * 32 + la;   // coalesced in la
        asm volatile("global_store_async_from_lds_b32 %0, %1, off"
                     :: "v"(gp), "v"(lds_off) : "memory");
      }
      asm volatile("s_wait_asynccnt 0x0" ::: "memory");
    }
#else
    for (int jj = 0; jj < 32; ++jj) {
      yrow[(size_t)jj * CH + ib * 32 + la] = myt[jj * 33 + la];
    }
#endif
    __syncthreads();
  }
}

// ---------------------------------------------------------------------------
// Launch: k1 (x_dbl WMMA) -> k2 (delta WMMA+softplus) -> k2b (BC transpose)
//         -> k3 (chunked scan).
// Workspace: x_dbl 6.3 MB + delta 25.2 MB + BC_t 4.2 MB  (~35.7 MB)
// ---------------------------------------------------------------------------
extern "C" void kernel_launch(void* const* d_in, const int* in_sizes, int n_in,
                              void* d_out, int out_size, void* d_ws, size_t ws_size,
                              hipStream_t stream) {
  const float* x     = (const float*)d_in[0];  // (B,K,D,L)
  const float* xpw   = (const float*)d_in[1];  // (K,44,D)
  const float* dtw   = (const float*)d_in[2];  // (K,D,R)
  const float* dtb   = (const float*)d_in[3];  // (K,D)
  const float* Alogs = (const float*)d_in[4];  // (K*D,N)
  const float* Dsv   = (const float*)d_in[5];  // (K*D,)
  float* yout = (float*)d_out;

  float* xdbl  = (float*)d_ws;                           // B*K*48*L floats
  float* delta = xdbl + (size_t)NB * NK * NCP * NL;      // B*K*D*L floats
  float* bct   = delta + (size_t)NB * NK * ND * NL;      // B*K*L*32 floats

  dim3 g1(NL / 16, NCP / 16, NB * NK);
  xdbl_wmma_kernel<<<g1, 32, 0, stream>>>(x, xpw, xdbl);

  dim3 g2(NL / 16, ND / 16, NB * NK);
  delta_wmma_kernel<<<g2, 32, 0, stream>>>(xdbl, dtw, dtb, delta);

  dim3 g2b(NL / 32, NB * NK);
  bct_transpose_kernel<<<g2b, 256, 0, stream>>>(xdbl, bct);

  dim3 g3(NB * NK * (ND / 8));
  scan_chunked_kernel<<<g3, 256, 0, stream>>>(x, delta, bct, Alogs, Dsv, yout);
}